// GCN_Encoder_19774029431051
// MI455X (gfx1250) — compile-verified
//
#include <hip/hip_runtime.h>

// ---------------- problem constants (match reference) ----------------
#define NN      50000       // N_NODES
#define NE      800000      // N_EDGES
#define D_IN    100         // CONCEPT_DIM
#define D_HID   200         // HIDDEN_DIM
#define D_OUT   100         // OUTPUT_DIM

typedef __attribute__((ext_vector_type(2))) float v2f;
typedef __attribute__((ext_vector_type(8))) float v8f;

// ---------------- zero ----------------
__global__ void zero_f32(float* __restrict__ p, long n) {
    long i = (long)blockIdx.x * blockDim.x + threadIdx.x;
    if (i < n) p[i] = 0.0f;
}

// ---------------- embedding gather: h0[n,f] = emb[ids[n], f] ----------------
__global__ void gather_emb(const float* __restrict__ emb,
                           const int* __restrict__ ids,
                           float* __restrict__ h0) {
    long i = (long)blockIdx.x * blockDim.x + threadIdx.x;
    const long total = (long)NN * D_IN;
    if (i >= total) return;
    int n = (int)(i / D_IN);
    int f = (int)(i - (long)n * D_IN);
    h0[i] = emb[(long)ids[n] * D_IN + f];
}

// ---------------- edge scatter-add: agg[dst] += h[src] ----------------
// One thread per (edge, feature). fp32 hardware atomics hit in L2 (node
// feature matrices are 20-40MB, well under the 192MB L2).
template <int D>
__global__ void scatter_add(const float* __restrict__ h,
                            const int* __restrict__ esrc,
                            const int* __restrict__ edst,
                            float* __restrict__ agg) {
    long tid = (long)blockIdx.x * blockDim.x + threadIdx.x;
    const long total = (long)NE * D;
    if (tid >= total) return;
    int e = (int)(tid / D);
    int f = (int)(tid - (long)e * D);
    int s = esrc[e];
    int d = edst[e];
    unsafeAtomicAdd(&agg[(long)d * D + f], h[(long)s * D + f]);
}

// ---------------- WMMA fp32 GEMM + bias + relu ----------------
// out[M,N] = relu(A[M,K] @ W[K,N] + bias[N])
// One wave per 16x16 output tile, K-loop in steps of 4 using
// V_WMMA_F32_16X16X4_F32. M must be a multiple of 16 (50000 = 3125*16),
// K a multiple of 4 (100, 200); N tail masked via clamp+scale so EXEC
// stays all-ones across the WMMA (ISA requirement).
__global__ void gemm_bias_relu_wmma(const float* __restrict__ A,
                                    const float* __restrict__ W,
                                    const float* __restrict__ bias,
                                    float* __restrict__ out,
                                    int M, int K, int N,
                                    int ntiles, int total_tiles) {
    int wave = (int)(((long)blockIdx.x * blockDim.x + threadIdx.x) >> 5);
    if (wave >= total_tiles) return;          // wave-uniform exit
    int lane = threadIdx.x & 31;

    int mt = wave / ntiles;
    int nt = wave - mt * ntiles;
    int m0 = mt << 4;
    int n0 = nt << 4;

    int half = lane >> 4;                     // 0: lanes 0-15, 1: lanes 16-31
    int l15  = lane & 15;

    // N-tail masking without touching EXEC
    int   ncol   = n0 + l15;
    int   ncc    = (ncol < N) ? ncol : (N - 1);
    float bscale = (ncol < N) ? 1.0f : 0.0f;

    // A row pointer for this lane: A[(m0+l15), k0 + 2*half + {0,1}]
    const float* arow = A + (long)(m0 + l15) * K + (half << 1);

    v8f c = {0.f, 0.f, 0.f, 0.f, 0.f, 0.f, 0.f, 0.f};

    for (int k0 = 0; k0 < K; k0 += 4) {
        v2f a;
        a.x = arow[k0];
        a.y = arow[k0 + 1];

        int kk = k0 + (half << 1);
        v2f b;
        b.x = W[(long)kk * N + ncc] * bscale;
        b.y = W[(long)(kk + 1) * N + ncc] * bscale;

        // D = A(16x4) * B(4x16) + C   -> v_wmma_f32_16x16x4_f32
        c = __builtin_amdgcn_wmma_f32_16x16x4_f32(
                /*neg_a=*/false, a, /*neg_b=*/false, b,
                /*c_mod=*/(short)0, c, /*reuse_a=*/false, /*reuse_b=*/false);
    }

    float bv = bias[ncc];
    if (ncol < N) {                           // divergent store OK (post-WMMA)
        #pragma unroll
        for (int r = 0; r < 8; ++r) {
            int row = m0 + r + (half << 3);   // VGPR r: M=r (lo half), M=8+r (hi)
            float v = c[r] + bv;
            out[(long)row * N + ncol] = v > 0.0f ? v : 0.0f;
        }
    }
}

// ---------------- host orchestration ----------------
extern "C" void kernel_launch(void* const* d_in, const int* in_sizes, int n_in,
                              void* d_out, int out_size, void* d_ws, size_t ws_size,
                              hipStream_t stream) {
    const float* emb  = (const float*)d_in[0];   // [VOCAB, 100]
    const float* W1   = (const float*)d_in[1];   // [100, 200]
    const float* b1   = (const float*)d_in[2];   // [200]
    const float* W2   = (const float*)d_in[3];   // [200, 100]
    const float* b2   = (const float*)d_in[4];   // [100]
    const int*   ids  = (const int*)d_in[5];     // [NN]
    const int*   esrc = (const int*)d_in[6];     // [NE]
    const int*   edst = (const int*)d_in[7];     // [NE]
    float*       out  = (float*)d_out;           // [NN, 100]

    float* ws   = (float*)d_ws;
    float* h0   = ws;                            // NN*100 = 5e6 floats
    float* agg1 = h0   + (long)NN * D_IN;        // 5e6
    float* h1   = agg1 + (long)NN * D_IN;        // NN*200 = 1e7
    float* agg2 = h1   + (long)NN * D_HID;       // 1e7   (total 120 MB)

    const int B = 256;
    long n_agg1 = (long)NN * D_IN;
    long n_agg2 = (long)NN * D_HID;
    long n_h0   = (long)NN * D_IN;

    // deterministic: re-zero accumulators every call
    zero_f32<<<(unsigned)((n_agg1 + B - 1) / B), B, 0, stream>>>(agg1, n_agg1);
    zero_f32<<<(unsigned)((n_agg2 + B - 1) / B), B, 0, stream>>>(agg2, n_agg2);

    // h0 = emb[ids]
    gather_emb<<<(unsigned)((n_h0 + B - 1) / B), B, 0, stream>>>(emb, ids, h0);

    // layer 1: agg1 = segment_sum(h0[src] -> dst)
    {
        long total = (long)NE * D_IN;
        scatter_add<D_IN><<<(unsigned)((total + B - 1) / B), B, 0, stream>>>(h0, esrc, edst, agg1);
    }
    // h1 = relu(agg1 @ W1 + b1)   M=50000 K=100 N=200
    {
        int ntiles = (D_HID + 15) / 16;                 // 13
        int total_tiles = (NN / 16) * ntiles;           // 3125*13
        int waves_per_blk = 128 / 32;
        int blocks = (total_tiles + waves_per_blk - 1) / waves_per_blk;
        gemm_bias_relu_wmma<<<blocks, 128, 0, stream>>>(agg1, W1, b1, h1,
                                                        NN, D_IN, D_HID,
                                                        ntiles, total_tiles);
    }
    // layer 2: agg2 = segment_sum(h1[src] -> dst)
    {
        long total = (long)NE * D_HID;
        scatter_add<D_HID><<<(unsigned)((total + B - 1) / B), B, 0, stream>>>(h1, esrc, edst, agg2);
    }
    // out = relu(agg2 @ W2 + b2)  M=50000 K=200 N=100
    {
        int ntiles = (D_OUT + 15) / 16;                 // 7
        int total_tiles = (NN / 16) * ntiles;           // 3125*7
        int waves_per_blk = 128 / 32;
        int blocks = (total_tiles + waves_per_blk - 1) / waves_per_blk;
        gemm_bias_relu_wmma<<<blocks, 128, 0, stream>>>(agg2, W2, b2, out,
                                                        NN, D_HID, D_OUT,
                                                        ntiles, total_tiles);
    }
}